// GlobalFilter_80822694576290
// MI455X (gfx1250) — compile-verified
//
#include <hip/hip_runtime.h>

// ---------------------------------------------------------------------------
// GFNet GlobalFilter on gfx1250 via fp32 WMMA (V_WMMA_F32_16X16X4_F32).
//
// rfft2(ortho) -> per-channel complex pointwise filter -> irfft2(ortho)
// factors into two channel-independent dense GEMMs over the 196 spatial
// samples with an elementwise complex rotation between them:
//   [yR;yI] = M1 * x        M1 = [P;Q], 224x196
//   zR = wR*yR - wI*yI ; zI = wI*yR + wR*yI          (per bin, per channel)
//   out = [VR,VI] * [zR;zI]  M2, 196x224
// with P[k,s]=cos(th)/14, Q=-sin(th)/14, VR[s,k]=(m_v/14)cos(th),
// VI=-(m_v/14)sin(th), th=2*pi*(u*p+v*q)/14, m_v=1 for v in {0,7} else 2.
//
// X tiles are staged into LDS with the gfx1250 async Global->LDS path
// (GLOBAL_LOAD_ASYNC_TO_LDS_B128, ASYNCcnt) when the builtin is available.
// ---------------------------------------------------------------------------

typedef float v2f __attribute__((ext_vector_type(2)));
typedef float v8f __attribute__((ext_vector_type(8)));
typedef int   v4i __attribute__((vector_size(16)));   // matches builtin param

#define AS1 __attribute__((address_space(1)))
#define AS3 __attribute__((address_space(3)))

#define A_DIM   14
#define BINS    112           // 14 * 8 rfft2 half-spectrum bins
#define S_DIM   196           // 14 * 14 spatial samples
#define CH      768
#define BATCH   256
#define COLS    64            // (batch,channel) columns per workgroup
#define LSTR    76            // LDS row stride in floats (bank-conflict tuned)
#define M1_ROWS 256           // 224 real rows padded to 16 row-tiles
#define M1_COLS 196
#define M2_ROWS 256           // 196 real rows padded to 16 row-tiles
#define M2_COLS 224

#if defined(__has_builtin)
#if __has_builtin(__builtin_amdgcn_global_load_async_to_lds_b128)
#define USE_ASYNC_LDS 1
#endif
#endif

__device__ __forceinline__ v8f wmma_f32(v2f a, v2f b, v8f c) {
  // D = A(16x4,f32) * B(4x16,f32) + C(16x16,f32)
  return __builtin_amdgcn_wmma_f32_16x16x4_f32(
      false, a, false, b, (short)0, c, false, false);
}

// ---------------------------------------------------------------------------
// Precompute the forward (M1) and inverse (M2) transform matrices in ws.
// Angles reduced mod 14 before the sin/cos for exactness.
// ---------------------------------------------------------------------------
__global__ void gf_init_mats(float* __restrict__ M1, float* __restrict__ M2) {
  const int idx = blockIdx.x * blockDim.x + threadIdx.x;
  const float TWO_PI_OV14 = 0.448798950512827605495f; // 2*pi/14

  if (idx < M1_ROWS * M1_COLS) {
    const int r = idx / M1_COLS, s = idx % M1_COLS;
    float val = 0.0f;
    if (r < 2 * BINS) {
      const int k = (r >= BINS) ? (r - BINS) : r;
      const int u = k >> 3, v = k & 7;
      const int p = s / A_DIM, q = s % A_DIM;
      const int t = (u * p + v * q) % A_DIM;
      const float ang = TWO_PI_OV14 * (float)t;
      val = ((r >= BINS) ? -sinf(ang) : cosf(ang)) * (1.0f / 14.0f);
    }
    M1[idx] = val;
  }

  if (idx < M2_ROWS * M2_COLS) {
    const int r = idx / M2_COLS, col = idx % M2_COLS;
    float val = 0.0f;
    if (r < S_DIM) {
      const int k = (col >= BINS) ? (col - BINS) : col;
      const int u = k >> 3, v = k & 7;
      const int p = r / A_DIM, q = r % A_DIM;
      const int t = (u * p + v * q) % A_DIM;
      const float ang = TWO_PI_OV14 * (float)t;
      const float mv = (v == 0 || v == 7) ? 1.0f : 2.0f;
      val = ((col >= BINS) ? -sinf(ang) : cosf(ang)) * (mv / 14.0f);
    }
    M2[idx] = val;
  }
}

// ---------------------------------------------------------------------------
// Fused filter kernel: one workgroup = one batch x 64 channels.
// 8 waves; each wave owns 2 row-tiles x 4 col-tiles (16x16) per GEMM.
// ---------------------------------------------------------------------------
__global__ __launch_bounds__(256) void gf_filter_wmma(
    const float* __restrict__ x,    // (256, 196, 768)
    const float* __restrict__ cw,   // (14, 8, 768, 2)
    const float* __restrict__ M1,   // (256, 196) row-major
    const float* __restrict__ M2,   // (256, 224) row-major
    float* __restrict__ out) {      // (256, 196, 768)
  extern __shared__ float smem[];
  float* Xs = smem;                 // [224][LSTR]: X tile, reused as filtered Z
  float* Ts = smem + 224 * LSTR;    // [224][LSTR]: raw spectrum [yR;yI]

  const int cb = blockIdx.x;        // channel block 0..11
  const int b  = blockIdx.y;        // batch 0..255
  const int chanBase = cb * COLS;
  const float* xb = x   + (size_t)b * S_DIM * CH + chanBase;
  float*       ob = out + (size_t)b * S_DIM * CH + chanBase;

  // ---- stage 0: stage X tile (196 x 64) into LDS ----
#ifdef USE_ASYNC_LDS
  {
    // Async DMA path: 16B chunks, 16 per 64-float row; ASYNCcnt-tracked,
    // no VGPR round trip. Both sides 16B-aligned (3072B / 304B row strides).
    AS3 float* XsL = (AS3 float*)Xs;
    for (int i = threadIdx.x; i < S_DIM * 16; i += 256) {
      const int s = i >> 4, nq = i & 15;
      __builtin_amdgcn_global_load_async_to_lds_b128(
          (AS1 v4i*)(xb + s * CH + nq * 4),
          (AS3 v4i*)(XsL + s * LSTR + nq * 4),
          0, 0);
    }
#if __has_builtin(__builtin_amdgcn_s_wait_asynccnt)
    __builtin_amdgcn_s_wait_asynccnt(0);
#else
    asm volatile("s_wait_asynccnt 0" ::: "memory");
#endif
  }
#else
  for (int i = threadIdx.x; i < S_DIM * COLS; i += 256) {
    const int s = i >> 6, n = i & 63;
    Xs[s * LSTR + n] = xb[s * CH + n];
  }
#endif
  __syncthreads();

  const int lane = threadIdx.x & 31;
  const int wave = threadIdx.x >> 5;
  const int m    = lane & 15;       // A row-in-tile, B/C column-in-tile
  const int kq   = lane >> 4;       // lane half: K pair select / +8 row select

  const v8f vzero = {0, 0, 0, 0, 0, 0, 0, 0};
  v8f acc[2][4];
#pragma unroll
  for (int i = 0; i < 2; ++i)
#pragma unroll
    for (int j = 0; j < 4; ++j) acc[i][j] = vzero;

  // ---- stage 1: spectrum = M1(256x196) * Xs(196x64) ----
  {
    const float* A0 = M1 + (size_t)((2 * wave + 0) * 16 + m) * M1_COLS;
    const float* A1 = M1 + (size_t)((2 * wave + 1) * 16 + m) * M1_COLS;
    for (int k = 0; k < M1_COLS; k += 4) {
      const v2f a0 = *(const v2f*)(A0 + k + 2 * kq);
      const v2f a1 = *(const v2f*)(A1 + k + 2 * kq);
      v2f bf[4];
#pragma unroll
      for (int ct = 0; ct < 4; ++ct) {
        const int n = ct * 16 + m;
        bf[ct].x = Xs[(k + 2 * kq + 0) * LSTR + n];
        bf[ct].y = Xs[(k + 2 * kq + 1) * LSTR + n];
      }
#pragma unroll
      for (int ct = 0; ct < 4; ++ct) {
        acc[0][ct] = wmma_f32(a0, bf[ct], acc[0][ct]);
        acc[1][ct] = wmma_f32(a1, bf[ct], acc[1][ct]);
      }
    }
  }

  // store spectrum (rows 0..223; padded row-tiles 14,15 are all-zero, skipped)
#pragma unroll
  for (int i = 0; i < 2; ++i) {
    const int rt = 2 * wave + i;
    if (rt < 14) {
#pragma unroll
      for (int ct = 0; ct < 4; ++ct)
#pragma unroll
        for (int j = 0; j < 8; ++j)
          Ts[(rt * 16 + j + 8 * kq) * LSTR + ct * 16 + m] = acc[i][ct][j];
    }
  }
  __syncthreads();

  // ---- stage 2: per-channel complex pointwise filter ----
  for (int i = threadIdx.x; i < BINS * COLS; i += 256) {
    const int k = i >> 6, n = i & 63;
    const float yR = Ts[k * LSTR + n];
    const float yI = Ts[(BINS + k) * LSTR + n];
    const float* w = cw + ((size_t)k * CH + chanBase + n) * 2;
    const float wR = w[0], wI = w[1];
    Xs[k * LSTR + n]          = yR * wR - yI * wI;
    Xs[(BINS + k) * LSTR + n] = yR * wI + yI * wR;
  }
  __syncthreads();

#pragma unroll
  for (int i = 0; i < 2; ++i)
#pragma unroll
    for (int j = 0; j < 4; ++j) acc[i][j] = vzero;

  // ---- stage 3: out = M2(256x224) * Z(224x64) ----
  {
    const float* A0 = M2 + (size_t)((2 * wave + 0) * 16 + m) * M2_COLS;
    const float* A1 = M2 + (size_t)((2 * wave + 1) * 16 + m) * M2_COLS;
    for (int k = 0; k < M2_COLS; k += 4) {
      const v2f a0 = *(const v2f*)(A0 + k + 2 * kq);
      const v2f a1 = *(const v2f*)(A1 + k + 2 * kq);
      v2f bf[4];
#pragma unroll
      for (int ct = 0; ct < 4; ++ct) {
        const int n = ct * 16 + m;
        bf[ct].x = Xs[(k + 2 * kq + 0) * LSTR + n];
        bf[ct].y = Xs[(k + 2 * kq + 1) * LSTR + n];
      }
#pragma unroll
      for (int ct = 0; ct < 4; ++ct) {
        acc[0][ct] = wmma_f32(a0, bf[ct], acc[0][ct]);
        acc[1][ct] = wmma_f32(a1, bf[ct], acc[1][ct]);
      }
    }
  }

  // ---- store output rows < 196, coalesced 16-float runs ----
#pragma unroll
  for (int i = 0; i < 2; ++i) {
    const int rt = 2 * wave + i;
#pragma unroll
    for (int ct = 0; ct < 4; ++ct)
#pragma unroll
      for (int j = 0; j < 8; ++j) {
        const int row = rt * 16 + j + 8 * kq;
        if (row < S_DIM)
          ob[(size_t)row * CH + ct * 16 + m] = acc[i][ct][j];
      }
  }
}

extern "C" void kernel_launch(void* const* d_in, const int* in_sizes, int n_in,
                              void* d_out, int out_size, void* d_ws, size_t ws_size,
                              hipStream_t stream) {
  const float* x  = (const float*)d_in[0];  // (256, 196, 768) fp32
  const float* cw = (const float*)d_in[1];  // (14, 8, 768, 2) fp32
  float* out = (float*)d_out;

  float* M1 = (float*)d_ws;                       // 256*196 floats
  float* M2 = M1 + (size_t)M1_ROWS * M1_COLS;     // 256*224 floats

  const int nInit = M2_ROWS * M2_COLS;            // 57344 >= 256*196
  gf_init_mats<<<(nInit + 255) / 256, 256, 0, stream>>>(M1, M2);

  dim3 grid(CH / COLS, BATCH);                    // (12, 256)
  const size_t smemBytes = 2u * 224u * LSTR * sizeof(float); // 136192 B
  gf_filter_wmma<<<grid, 256, smemBytes, stream>>>(x, cw, M1, M2, out);
}